// ConsistencyLoss_64991445123088
// MI455X (gfx1250) — compile-verified
//
#include <hip/hip_runtime.h>

// CDNA5 / gfx1250: wave32, WMMA 16x16x4 F32 (K=4, pad xyz with 0).
typedef __attribute__((ext_vector_type(2))) float v2f;
typedef __attribute__((ext_vector_type(8))) float v8f;

#define B_   8
#define NP_  2048
#define NC_  8192
#define KNN_ 16
#define TILE 16

// D = A(16x4) x B(4x16) + 0  -> 16x16 inner products in one v_wmma_f32_16x16x4_f32.
// A layout (32-bit A 16x4): lanes 0-15: M=lane, a.x=K0, a.y=K1 ; lanes 16-31: M=lane-16, a.x=K2, a.y=K3
// B layout (4x16, 2 VGPRs): lanes 0-15: N=lane, b.x=K0, b.y=K1 ; lanes 16-31: N=lane-16, b.x=K2, b.y=K3
// C/D layout: VGPR v: lanes 0-15 -> (M=v, N=lane); lanes 16-31 -> (M=v+8, N=lane-16)
__device__ __forceinline__ v8f wmma_inner(v2f a, v2f b) {
  v8f c = {};
  return __builtin_amdgcn_wmma_f32_16x16x4_f32(
      /*neg_a=*/false, a, /*neg_b=*/false, b,
      /*c_mod=*/(short)0, c, /*reuse_a=*/false, /*reuse_b=*/false);
}

// ---------------------------------------------------------------------------
// Kernel 1: for each partial point, argmin distance over completed (8192),
// gather matched point. One wave per 16-row tile. Row term ||p||^2 cancels
// in argmin, so score = ||c_j||^2 - 2 <p_i, c_j>.
// ---------------------------------------------------------------------------
__global__ __launch_bounds__(32) void k_chamfer_match(
    const float* __restrict__ completed, const float* __restrict__ partial,
    float* __restrict__ matched) {
  const int b    = blockIdx.y;
  const int m0   = blockIdx.x * TILE;
  const int lane = threadIdx.x;
  const int col  = lane & 15;
  const bool hi  = lane >= 16;

  const float* Pb = partial   + (size_t)b * NP_ * 3;
  const float* Cb = completed + (size_t)b * NC_ * 3;

  // A tile (rows m0..m0+15 of partial), K padded to 4 with 0
  const float* pr = Pb + (m0 + col) * 3;
  const float px = pr[0], py = pr[1], pz = pr[2];
  v2f a; a.x = hi ? pz : px; a.y = hi ? 0.f : py;

  float bd[8]; int bi[8];
#pragma unroll
  for (int v = 0; v < 8; ++v) { bd[v] = 3.0e38f; bi[v] = 0; }

  for (int n0 = 0; n0 < NC_; n0 += TILE) {
    const float* cp = Cb + (n0 + col) * 3;
    __builtin_prefetch(cp + 2 * TILE * 3, 0, 1);   // global_prefetch_b8, 2 tiles ahead
    const float cx = cp[0], cy = cp[1], cz = cp[2];
    const float sqc = cx * cx + cy * cy + cz * cz;
    v2f bv; bv.x = hi ? cz : cx; bv.y = hi ? 0.f : cy;
    const v8f d = wmma_inner(a, bv);
    const int j = n0 + col;
#pragma unroll
    for (int v = 0; v < 8; ++v) {
      const float s = sqc - 2.0f * d[v];
      if (s < bd[v]) { bd[v] = s; bi[v] = j; }   // strict < keeps earliest index in-lane
    }
  }

  // Reduce (min, argmin) across the 16 lanes of each half (xor stays in-half).
#pragma unroll
  for (int v = 0; v < 8; ++v) {
#pragma unroll
    for (int off = 1; off < 16; off <<= 1) {
      const float od = __shfl_xor(bd[v], off, 32);
      const int   oi = __shfl_xor(bi[v], off, 32);
      if (od < bd[v] || (od == bd[v] && oi < bi[v])) { bd[v] = od; bi[v] = oi; }
    }
  }

  if (lane == 0 || lane == 16) {
    const int rbase = hi ? 8 : 0;
#pragma unroll
    for (int v = 0; v < 8; ++v) {
      const int row = m0 + rbase + v;
      const float* src = Cb + (size_t)bi[v] * 3;
      float* dst = matched + ((size_t)b * NP_ + row) * 3;
      dst[0] = src[0]; dst[1] = src[1]; dst[2] = src[2];
    }
  }
}

// ---------------------------------------------------------------------------
// Kernel 2: self-kNN (k=16, self included) + local geometry features.
// 4 waves / 16-row tile. Distances staged in LDS chunks via WMMA; 16 selector
// threads keep a register-resident, fully unrolled, stable sorted top-16.
// MODE 0: write partial features. MODE 1: diff vs featP, accumulate SSE.
// ---------------------------------------------------------------------------
template <int MODE>
__global__ __launch_bounds__(128) void k_knn_feat(
    const float* __restrict__ X,       // (B, NP, 3)
    const float* __restrict__ featP,   // read when MODE==1
    float* __restrict__ featOut,       // write when MODE==0
    float* __restrict__ psum) {        // write when MODE==1
  const int b    = blockIdx.y;
  const int m0   = blockIdx.x * TILE;
  const int tid  = threadIdx.x;
  const int wave = tid >> 5;
  const int lane = tid & 31;
  const int col  = lane & 15;
  const bool hi  = lane >= 16;

  constexpr int CHUNK = 512;
  constexpr int LD    = CHUNK + 8;      // pad LDS rows -> no 16-way bank conflict
  __shared__ float sdist[TILE * LD];    // ~33 KB of the 320 KB WGP LDS
  __shared__ float sred[128];

  const float* Xb = X + (size_t)b * NP_ * 3;

  const float* pr = Xb + (m0 + col) * 3;
  v2f a; { const float px = pr[0], py = pr[1], pz = pr[2];
           a.x = hi ? pz : px; a.y = hi ? 0.f : py; }

  float kd[KNN_]; int ki[KNN_];
#pragma unroll
  for (int q = 0; q < KNN_; ++q) { kd[q] = 3.0e38f; ki[q] = 0; }

  for (int chunk = 0; chunk < NP_ / CHUNK; ++chunk) {
    const int cbase = chunk * CHUNK;
    for (int it = 0; it < CHUNK / (4 * TILE); ++it) {
      const int n0 = cbase + (it * 4 + wave) * TILE;
      const float* cp = Xb + (n0 + col) * 3;
      const float cx = cp[0], cy = cp[1], cz = cp[2];
      const float sqc = cx * cx + cy * cy + cz * cz;
      v2f bv; bv.x = hi ? cz : cx; bv.y = hi ? 0.f : cy;
      const v8f d = wmma_inner(a, bv);
      const int lcol = (n0 - cbase) + col;
#pragma unroll
      for (int v = 0; v < 8; ++v) {
        const int row = (hi ? 8 : 0) + v;
        sdist[row * LD + lcol] = sqc - 2.0f * d[v];  // row-constant term cancels in top-k
      }
    }
    __syncthreads();
    if (tid < TILE) {
      for (int c = 0; c < CHUNK; ++c) {
        const float s = sdist[tid * LD + c];
        if (s < kd[KNN_ - 1]) {          // strict < : stable, earlier index wins ties
          kd[KNN_ - 1] = s; ki[KNN_ - 1] = cbase + c;
#pragma unroll
          for (int q = KNN_ - 1; q > 0; --q) {   // bubble new element up (registers only)
            if (kd[q] < kd[q - 1]) {
              const float td = kd[q]; kd[q] = kd[q - 1]; kd[q - 1] = td;
              const int   ti = ki[q]; ki[q] = ki[q - 1]; ki[q - 1] = ti;
            }
          }
        }
      }
    }
    __syncthreads();
  }

  float acc = 0.f;
  if (tid < TILE) {
    const int gi = m0 + tid;
    const float* p = Xb + (size_t)gi * 3;
    const float px = p[0], py = p[1], pz = p[2];
#pragma unroll
    for (int j = 0; j < KNN_; ++j) {
      const float* nb = Xb + (size_t)ki[j] * 3;
      const float rx = nb[0] - px, ry = nb[1] - py, rz = nb[2] - pz;
      const size_t fo = ((size_t)b * NP_ + gi) * (KNN_ * 3) + j * 3;
      if (MODE == 0) {
        featOut[fo + 0] = rx; featOut[fo + 1] = ry; featOut[fo + 2] = rz;
      } else {
        const float d0 = rx - featP[fo + 0];
        const float d1 = ry - featP[fo + 1];
        const float d2 = rz - featP[fo + 2];
        acc += d0 * d0 + d1 * d1 + d2 * d2;
      }
    }
  }
  if (MODE == 1) {
    sred[tid] = acc;
    __syncthreads();
    if (tid == 0) {
      float s = 0.f;
      for (int t = 0; t < 128; ++t) s += sred[t];
      psum[blockIdx.y * gridDim.x + blockIdx.x] = s;
    }
  }
}

// ---------------------------------------------------------------------------
// Kernel 3: deterministic final reduction -> WEIGHT * mean.
// ---------------------------------------------------------------------------
__global__ __launch_bounds__(256) void k_final(const float* __restrict__ psum, int n,
                                               float* __restrict__ out) {
  __shared__ float sred[256];
  float s = 0.f;
  for (int i = threadIdx.x; i < n; i += 256) s += psum[i];
  sred[threadIdx.x] = s;
  __syncthreads();
  for (int off = 128; off > 0; off >>= 1) {
    if (threadIdx.x < off) sred[threadIdx.x] += sred[threadIdx.x + off];
    __syncthreads();
  }
  if (threadIdx.x == 0)
    out[0] = 1.0f /*WEIGHT*/ * sred[0] / (float)((size_t)B_ * NP_ * KNN_ * 3);
}

// ---------------------------------------------------------------------------
extern "C" void kernel_launch(void* const* d_in, const int* in_sizes, int n_in,
                              void* d_out, int out_size, void* d_ws, size_t ws_size,
                              hipStream_t stream) {
  const float* completed = (const float*)d_in[0];   // (8, 8192, 3) f32
  const float* partial   = (const float*)d_in[1];   // (8, 2048, 3) f32
  float* out = (float*)d_out;                       // scalar f32

  float* ws      = (float*)d_ws;
  float* matched = ws;                                   // B*NP*3   =  49152 f
  float* featP   = matched + (size_t)B_ * NP_ * 3;       // B*NP*48  = 786432 f
  float* psum    = featP   + (size_t)B_ * NP_ * 48;      // B*(NP/16)=   1024 f

  const dim3 grid(NP_ / TILE, B_);
  k_chamfer_match<<<grid, 32, 0, stream>>>(completed, partial, matched);
  k_knn_feat<0><<<grid, 128, 0, stream>>>(partial, nullptr, featP, nullptr);
  k_knn_feat<1><<<grid, 128, 0, stream>>>(matched, featP, nullptr, psum);
  k_final<<<1, 256, 0, stream>>>(psum, B_ * (NP_ / TILE), out);
}